// DenseCapsule_12601434046794
// MI455X (gfx1250) — compile-verified
//
#include <hip/hip_runtime.h>

// Capsule routing, fused 3-pass design for MI455X (gfx1250, wave32).
// B=32, IN_N=2048, IN_D=16, OUT_N=64, OUT_D=32, ROUTINGS=3.
#define B_      32
#define IN_N_   2048
#define IN_D_   16
#define OUT_N_  64
#define OUT_D_  32
#define TI      16          // i-tile per workgroup
#define NTHREADS 512        // 16 waves (wave32)
#define O_PER_WAVE 4        // 16 waves * 4 = 64 output capsules

typedef __attribute__((ext_vector_type(2))) float v2f;
typedef __attribute__((ext_vector_type(8))) float v8f;

// Compute x_hat[:, o, i, :] = X_i (32x16) * W_oi^T (16x32) as four 16x16 f32
// WMMA tiles with K=16 (4 k-steps of V_WMMA_F32_16X16X4_F32).
// A 16x4 f32 layout: lane l holds {A[m, 2*(l>>4)+s*4], A[m, ...+1]}, m = l&15.
// B 4x16 f32 layout: lane l holds {B[2*(l>>4)+s*4, n], B[...+1, n]}, n = l&15.
// C/D 16x16 f32: vgpr r, lane l -> row r + 8*(l>>4), col l&15.
__device__ __forceinline__ void compute_xhat(
    const float* __restrict__ lds_x,   // [b][il][k] : (b*TI + il)*16 + k
    int il,
    const float* __restrict__ wblk,    // &W[o][i][0][0], layout [d][k], k stride 1
    int lane,
    v8f acc[2][2])
{
    const int half2 = (lane >> 4) * 2;
    const int l15   = lane & 15;
#pragma unroll
    for (int mt = 0; mt < 2; ++mt)
#pragma unroll
        for (int nt = 0; nt < 2; ++nt)
#pragma unroll
            for (int q = 0; q < 8; ++q) acc[mt][nt][q] = 0.0f;

#pragma unroll
    for (int s = 0; s < 4; ++s) {
        const int kk = s * 4 + half2;
        v2f a0 = *(const v2f*)&lds_x[((l15     ) * TI + il) * IN_D_ + kk];
        v2f a1 = *(const v2f*)&lds_x[((l15 + 16) * TI + il) * IN_D_ + kk];
        v2f b0 = *(const v2f*)&wblk[(l15     ) * IN_D_ + kk];
        v2f b1 = *(const v2f*)&wblk[(l15 + 16) * IN_D_ + kk];
        acc[0][0] = __builtin_amdgcn_wmma_f32_16x16x4_f32(false, a0, false, b0,
                        (short)0, acc[0][0], false, false);
        acc[0][1] = __builtin_amdgcn_wmma_f32_16x16x4_f32(false, a0, false, b1,
                        (short)0, acc[0][1], false, false);
        acc[1][0] = __builtin_amdgcn_wmma_f32_16x16x4_f32(false, a1, false, b0,
                        (short)0, acc[1][0], false, false);
        acc[1][1] = __builtin_amdgcn_wmma_f32_16x16x4_f32(false, a1, false, b1,
                        (short)0, acc[1][1], false, false);
    }
}

// One routing pass, fused: (iter>=1) logits + softmax over out_n in LDS, then
// c-weighted accumulation of s[b,o,d] (x_hat recomputed via WMMA both times,
// weight tile hits L2 on the second read).
__global__ __launch_bounds__(NTHREADS)
void caps_pass(const float* __restrict__ x,          // [B][IN_N][IN_D]
               const float* __restrict__ w,          // [OUT_N][IN_N][OUT_D][IN_D]
               const float* __restrict__ outs_prev,  // [B][OUT_N][OUT_D]
               float* __restrict__ b_buf,            // [B][OUT_N][IN_N]
               float* __restrict__ s_buf,            // [B][OUT_N][OUT_D] (atomic accum)
               int iter)
{
    __shared__ float lds_x[B_ * TI * IN_D_];     // 32 KB
    __shared__ float lds_c[B_ * OUT_N_ * TI];    // 128 KB: logits, then softmax c

    const int i0   = blockIdx.x * TI;
    const int wave = threadIdx.x >> 5;
    const int lane = threadIdx.x & 31;

    // Stage x tile into LDS with CDNA5 async copies (global -> LDS, no VGPR
    // round-trip, tracked by ASYNCcnt). Tile and LDS layouts are both
    // (b-major, then il*16+k) so the linear element index e maps 1:1.
    for (int v4 = threadIdx.x; v4 < (B_ * TI * IN_D_) / 4; v4 += NTHREADS) {
        const int e   = v4 * 4;
        const int b   = e / (TI * IN_D_);
        const int rem = e - b * (TI * IN_D_);
        const float* src = &x[(size_t)b * IN_N_ * IN_D_ + (size_t)i0 * IN_D_ + rem];
        const uint32_t dst = (uint32_t)(uintptr_t)&lds_x[e];  // flat low 32 = LDS offset
        asm volatile("global_load_async_to_lds_b128 %0, %1, off"
                     :: "v"(dst), "v"(src) : "memory");
    }
    asm volatile("s_wait_asynccnt 0x0" ::: "memory");
    __syncthreads();

    if (iter >= 1) {
        // ---- Phase 1: raw agreement dot  p[b,o,i] = sum_d outs_prev * x_hat
        for (int j = 0; j < O_PER_WAVE; ++j) {
            const int o = wave * O_PER_WAVE + j;
            for (int il = 0; il < TI; ++il) {
                const int i = i0 + il;
                const float* wblk = w + ((size_t)o * IN_N_ + i) * (OUT_D_ * IN_D_);
                v8f acc[2][2];
                compute_xhat(lds_x, il, wblk, lane, acc);
                const int d0 = lane & 15;
#pragma unroll
                for (int mt = 0; mt < 2; ++mt) {
#pragma unroll
                    for (int r = 0; r < 8; ++r) {
                        const int bb = mt * 16 + r + 8 * (lane >> 4);
                        const float* op = &outs_prev[((size_t)bb * OUT_N_ + o) * OUT_D_];
                        float p = acc[mt][0][r] * op[d0] + acc[mt][1][r] * op[d0 + 16];
                        // reduce over the 16 lanes holding d
                        p += __shfl_xor(p, 1);
                        p += __shfl_xor(p, 2);
                        p += __shfl_xor(p, 4);
                        p += __shfl_xor(p, 8);
                        if (d0 == 0)
                            lds_c[(bb * OUT_N_ + o) * TI + il] = p;
                    }
                }
            }
        }
        __syncthreads();

        // ---- Phase 2: add b_prev (coalesced, outside the WMMA loop), persist
        //      logits (iter==1 only: b1 = 0 + dot), softmax over out_n per (b,i).
        {
            const int b  = threadIdx.x >> 4;   // 0..31
            const int il = threadIdx.x & 15;   // 0..15
            const int i  = i0 + il;
            float* row = &lds_c[(b * OUT_N_) * TI + il];  // stride TI per o
            float m = -3.4e38f;
            for (int o = 0; o < OUT_N_; ++o) {
                float v = row[o * TI];
                if (iter >= 2)
                    v += b_buf[((size_t)b * OUT_N_ + o) * IN_N_ + i];
                else
                    b_buf[((size_t)b * OUT_N_ + o) * IN_N_ + i] = v;
                row[o * TI] = v;
                m = fmaxf(m, v);
            }
            float ssum = 0.0f;
            for (int o = 0; o < OUT_N_; ++o) {
                float e = __expf(row[o * TI] - m);
                ssum += e;
                row[o * TI] = e;
            }
            const float inv = 1.0f / ssum;
            for (int o = 0; o < OUT_N_; ++o) row[o * TI] *= inv;
        }
        __syncthreads();
    }

    // ---- Phase 3: s[b,o,d] += sum_i c[b,o,i] * x_hat[b,o,i,d]
    for (int j = 0; j < O_PER_WAVE; ++j) {
        const int o = wave * O_PER_WAVE + j;
        v8f sacc[2][2];
#pragma unroll
        for (int mt = 0; mt < 2; ++mt)
#pragma unroll
            for (int nt = 0; nt < 2; ++nt)
#pragma unroll
                for (int q = 0; q < 8; ++q) sacc[mt][nt][q] = 0.0f;

        for (int il = 0; il < TI; ++il) {
            const int i = i0 + il;
            const float* wblk = w + ((size_t)o * IN_N_ + i) * (OUT_D_ * IN_D_);
            if (il + 1 < TI)  // prefetch next 2KB weight block (one line per lane)
                __builtin_prefetch(wblk + OUT_D_ * IN_D_ + lane * 16, 0, 1);
            v8f acc[2][2];
            compute_xhat(lds_x, il, wblk, lane, acc);
#pragma unroll
            for (int mt = 0; mt < 2; ++mt) {
#pragma unroll
                for (int r = 0; r < 8; ++r) {
                    const int bb = mt * 16 + r + 8 * (lane >> 4);
                    const float cv = (iter == 0)
                        ? 0.015625f                       // softmax of zeros = 1/64
                        : lds_c[(bb * OUT_N_ + o) * TI + il];
                    sacc[mt][0][r] += cv * acc[mt][0][r];
                    sacc[mt][1][r] += cv * acc[mt][1][r];
                }
            }
        }
        const int d0 = lane & 15;
#pragma unroll
        for (int mt = 0; mt < 2; ++mt)
#pragma unroll
            for (int nt = 0; nt < 2; ++nt)
#pragma unroll
                for (int r = 0; r < 8; ++r) {
                    const int bb = mt * 16 + r + 8 * (lane >> 4);
                    const int dd = nt * 16 + d0;
                    unsafeAtomicAdd(&s_buf[((size_t)bb * OUT_N_ + o) * OUT_D_ + dd],
                                    sacc[mt][nt][r]);
                }
    }
}

// outputs[b,o,:] = squash(s[b,o,:]); one wave per (b,o) vector (OUT_D=32 lanes).
__global__ __launch_bounds__(256)
void squash_kernel(const float* __restrict__ s, float* __restrict__ out)
{
    const int v    = blockIdx.x * 8 + (threadIdx.x >> 5);
    const int lane = threadIdx.x & 31;
    float val = s[(size_t)v * OUT_D_ + lane];
    float sq = val * val;
    sq += __shfl_xor(sq, 1);
    sq += __shfl_xor(sq, 2);
    sq += __shfl_xor(sq, 4);
    sq += __shfl_xor(sq, 8);
    sq += __shfl_xor(sq, 16);
    const float n = sqrtf(sq);
    const float scale = sq / (1.0f + sq) / (n + 1e-8f);
    out[(size_t)v * OUT_D_ + lane] = scale * val;
}

extern "C" void kernel_launch(void* const* d_in, const int* in_sizes, int n_in,
                              void* d_out, int out_size, void* d_ws, size_t ws_size,
                              hipStream_t stream)
{
    const float* x = (const float*)d_in[0];   // [32, 2048, 16]
    const float* w = (const float*)d_in[1];   // [64, 2048, 32, 16]
    float* out = (float*)d_out;               // [32, 64, 32]

    // Workspace layout (floats): s accumulator | outputs | routing logits b
    float* s_buf    = (float*)d_ws;                            // 65536
    float* outs_buf = s_buf + (size_t)B_ * OUT_N_ * OUT_D_;    // 65536
    float* b_buf    = outs_buf + (size_t)B_ * OUT_N_ * OUT_D_; // 4M

    const dim3 pg(IN_N_ / TI), pb(NTHREADS);
    const int nvec = B_ * OUT_N_;  // 2048 (b,o) vectors

    for (int iter = 0; iter < 3; ++iter) {
        hipMemsetAsync(s_buf, 0, (size_t)B_ * OUT_N_ * OUT_D_ * sizeof(float), stream);
        caps_pass<<<pg, pb, 0, stream>>>(x, w, outs_buf, b_buf, s_buf, iter);
        float* dst = (iter == 2) ? out : outs_buf;
        squash_kernel<<<nvec / 8, 256, 0, stream>>>(s_buf, dst);
    }
}